// UnifiedPolypLoss_51625506898071
// MI455X (gfx1250) — compile-verified
//
#include <hip/hip_runtime.h>
#include <hip/hip_bf16.h>

// UnifiedPolypLoss fused kernel for MI455X (gfx1250, wave32).
//
// Strategy (memory-roofline ~5.5us for 128MiB):
//  - grid (16 tiles, 64 images); each block = 256 threads owns a 64x256 tile.
//  - TDM tensor_load_to_lds stages the target tile + halo (94 x 271 f32,
//    global stride 512, LDS stride 288 via iterate mode) in ONE instruction,
//    tracked by TENSORcnt (s_wait_tensorcnt in the issuing wave only).
//  - horizontal 31-sums (u16, exact: target is binary) + per-column integer
//    sliding window => 31x31 box sum at ~5 ops/pixel.
//  - separable Sobel with rolling row partials (exact small ints).
//  - ONE log per pixel: bce = -log(t ? p : 1-p) since t in {0,1}.
//  - deterministic: per-tile partials to d_ws, fixed-order double reduction
//    in a tiny finalize kernel (no floating-point atomics).

#define Hh 512
#define Ww 512
#define TRr 64          // tile rows
#define TCc 256         // tile cols
#define TLROWS 94       // TRr + 2*15
#define TLS 288         // LDS row stride (floats) for target tile
#define HSS 264         // LDS row stride (u16) for horizontal sums
#define COPYW 271       // valid copy width per tile (both c0=0 and c0=256)

typedef unsigned int u32x4 __attribute__((ext_vector_type(4)));
typedef int i32x4 __attribute__((ext_vector_type(4)));
typedef int i32x8 __attribute__((ext_vector_type(8)));

// Device-pass-only probe: if the gfx1250 compiler lacks the TDM builtin this
// fails with "compiling for gfx1250" (the round-2 failure was the HOST pass,
// which never has AMDGCN builtins — hence the __HIP_DEVICE_COMPILE__ scoping).
#if defined(__HIP_DEVICE_COMPILE__) && !__has_builtin(__builtin_amdgcn_tensor_load_to_lds)
#error CDNA5_PROBE_NO_TDM_BUILTIN_ON_DEVICE
#endif

#if defined(__HIP_DEVICE_COMPILE__)
__device__ __forceinline__ unsigned lds_offset_of(const void* p) {
  // addrspacecast generic -> LDS(3), then ptrtoint => 32-bit LDS byte offset
  return (unsigned)(unsigned long long)(__attribute__((address_space(3))) const void*)p;
}
#endif

__global__ __launch_bounds__(256) void polyp_main(
    const float* __restrict__ pred, const float* __restrict__ target,
    float* __restrict__ parts) {
  __shared__ __align__(16) float TL[TLROWS * TLS];            // 108,288 B
  __shared__ __align__(16) unsigned short HS[TLROWS * HSS];   //  49,632 B

  const int tid = threadIdx.x;
  const int tile = blockIdx.x;            // 0..15
  const int b = blockIdx.y;               // 0..63
  const int c0 = (tile & 1) * TCc;        // 0 or 256
  const int r0 = (tile >> 1) * TRr;       // 0,64,...,448

  // ---- phase 0: zero the halo tile (handles top/left/right/bottom padding)
  float4 z4 = make_float4(0.f, 0.f, 0.f, 0.f);
  float4* TL4 = (float4*)TL;
  for (int i = tid; i < (TLROWS * TLS) / 4; i += 256) TL4[i] = z4;
  __syncthreads();

  // ---- phase 1: stage valid sub-rectangle of target into LDS via TDM
  const int y_lo = r0 - 15;
  const int y_start = y_lo < 0 ? 0 : y_lo;
  const int y_end = (r0 + 79) > Hh ? Hh : (r0 + 79);
  const int tile_h = y_end - y_start;                  // 79 or 94
  const int lds_row = y_start - y_lo;                  // 0 or 15
  const int x_start = (c0 == 0) ? 0 : (c0 - 15);
  const int lds_col = (c0 == 0) ? 15 : 0;
  const float* gsrc = target + ((size_t)(b * Hh + y_start)) * Ww + x_start;

#if defined(__HIP_DEVICE_COMPILE__)
  if (tid < 32) {  // one wave issues the TDM op (EXEC is ignored; issue once)
    unsigned ldsa = lds_offset_of(&TL[lds_row * TLS + lds_col]);
    unsigned long long ga = (unsigned long long)gsrc;
    // D# group0: count=1, lds_addr, 57b global addr, type=2
    u32x4 g0 = {1u, ldsa, (unsigned)(ga & 0xFFFFFFFFu),
                (unsigned)(((ga >> 32) & 0x01FFFFFFu) | 0x80000000u)};
    // D# group1: data_size=4B (code 2), iterate_enable; dims sized exactly to
    // the clamped rectangle so no OOB path is exercised.
    i32x8 g1 = {(int)((2u << 16) | (1u << 19)),
                (int)(COPYW << 16),            // tensor_dim0 lo16
                (int)(tile_h << 16),           // dim0 hi16=0 | tensor_dim1 lo16
                (int)(COPYW << 16),            // dim1 hi16=0 | tile_dim0
                0,                             // tile_dim1=0, tile_dim2=0
                Ww,                            // tensor_dim0_stride = 512
                0, 0};
    // D# group2 (iterate mode): lds_inc=288, global_inc=512, count=tile_h-1
    i32x4 g2 = {0, TLS, Ww, (int)((tile_h - 1) << 16)};
    i32x4 g3 = {0, 0, 0, 0};
#if __has_include(<hip/amd_detail/amd_gfx1250_TDM.h>)
    i32x8 g4 = {0, 0, 0, 0, 0, 0, 0, 0};
    __builtin_amdgcn_tensor_load_to_lds(g0, g1, g2, g3, g4, 0);
#else
    __builtin_amdgcn_tensor_load_to_lds(g0, g1, g2, g3, 0);
#endif
#if __has_builtin(__builtin_amdgcn_s_wait_tensorcnt)
    __builtin_amdgcn_s_wait_tensorcnt(0);
#else
    asm volatile("s_wait_tensorcnt 0" ::: "memory");
#endif
  }
#endif  // __HIP_DEVICE_COMPILE__ (host pass only parses this kernel)
  __syncthreads();

  // ---- phase 2: horizontal sliding 31-sums -> u16 (exact, values <= 31)
  for (int idx = tid; idx < TLROWS * 4; idx += 256) {
    const int seg = idx / TLROWS;          // 0..3 (seg-major: fewer conflicts)
    const int j = idx - seg * TLROWS;      // 0..93
    const int cb = seg * 64;
    const float* row = TL + j * TLS + cb;
    float s = 0.f;
#pragma unroll
    for (int u = 0; u < 30; ++u) s += row[u];
    unsigned short* hrow = HS + j * HSS + cb;
    for (int k = 0; k < 64; ++k) {
      s += row[k + 30];
      hrow[k] = (unsigned short)(int)(s + 0.5f);
      s -= row[k];
    }
  }
  __syncthreads();

  // ---- phase 3: per-column walk; vertical sliding sum + sobel + losses
  const int c = tid;  // block-local output column
  float rh0, rh1, rh2, rd0, rd1, rd2, t1c;
  {
    const float* ra = TL + 14 * TLS + c + 14;
    rh0 = ra[0] + 2.f * ra[1] + ra[2];
    rd0 = ra[2] - ra[0];
    const float* rb = TL + 15 * TLS + c + 14;
    rh1 = rb[0] + 2.f * rb[1] + rb[2];
    rd1 = rb[2] - rb[0];
    t1c = rb[1];
  }
  int vs = 0;
#pragma unroll
  for (int j = 0; j < 30; ++j) vs += HS[j * HSS + c];

  float acc_pt = 0.f, acc_p = 0.f, acc_t = 0.f, acc_bl = 0.f, acc_sl = 0.f;
  const float* pp = pred + ((size_t)(b * Hh + r0)) * Ww + c0 + c;
  for (int r = 0; r < TRr; ++r) {
    const float* rn = TL + (r + 16) * TLS + c + 14;
    rh2 = rn[0] + 2.f * rn[1] + rn[2];
    rd2 = rn[2] - rn[0];
    vs += HS[(r + 30) * HSS + c];
    float p = *pp;
    __builtin_prefetch(pp + 4 * Ww, 0, 0);  // stream-ahead: global_prefetch_b8
    float t = t1c;
    float gx = rd0 + 2.f * rd1 + rd2;
    float gy = rh2 - rh0;
    float q = (t > 0.5f) ? p : (1.0f - p);
    float bce = -__logf(q);                           // one trans op per pixel
    float mag = sqrtf(fmaf(gx, gx, fmaf(gy, gy, 1e-12f)));
    float bw = fmaf(10.f, mag, 1.f);
    float pooled = (float)vs * (1.0f / 961.0f);
    float wv = fmaf(5.f, fabsf(pooled - t), 1.f);
    acc_bl = fmaf(bw, bce, acc_bl);
    acc_sl = fmaf(wv, bce, acc_sl);
    acc_pt = fmaf(p, t, acc_pt);
    acc_p += p;
    acc_t += t;
    vs -= HS[r * HSS + c];
    rh0 = rh1; rh1 = rh2; rd0 = rd1; rd1 = rd2; t1c = rn[1];
    pp += Ww;
  }

  // ---- phase 4: block reduction (reuse HS as f32 scratch), write partials
  __syncthreads();
  float* red = (float*)HS;
  red[tid] = acc_pt;
  red[tid + 256] = acc_p;
  red[tid + 512] = acc_t;
  red[tid + 768] = acc_bl;
  red[tid + 1024] = acc_sl;
  __syncthreads();
  for (int s2 = 128; s2 > 0; s2 >>= 1) {
    if (tid < s2) {
      red[tid] += red[tid + s2];
      red[tid + 256] += red[tid + 256 + s2];
      red[tid + 512] += red[tid + 512 + s2];
      red[tid + 768] += red[tid + 768 + s2];
      red[tid + 1024] += red[tid + 1024 + s2];
    }
    __syncthreads();
  }
  if (tid == 0) {
    float* o = parts + ((size_t)(b * 16 + tile)) * 5;
    o[0] = red[0];
    o[1] = red[256];
    o[2] = red[512];
    o[3] = red[768];
    o[4] = red[1024];
  }
}

__global__ __launch_bounds__(64) void polyp_final(
    const float* __restrict__ parts, float* __restrict__ out) {
  __shared__ double sterm[64], sbl[64], ssl[64];
  const int b = threadIdx.x;  // 64 threads == 64 samples
  double s_pt = 0, s_p = 0, s_t = 0, s_bl = 0, s_sl = 0;
  for (int t = 0; t < 16; ++t) {
    const float* q = parts + ((size_t)(b * 16 + t)) * 5;
    s_pt += q[0]; s_p += q[1]; s_t += q[2]; s_bl += q[3]; s_sl += q[4];
  }
  const double TP = s_pt, FP = s_p - s_pt, FN = s_t - s_pt;
  const double tv = (TP + 1e-6) / (TP + 0.3 * FP + 0.7 * FN + 1e-6);
  double one_m = 1.0 - tv;
  if (one_m < 0.0) one_m = 0.0;
  sterm[b] = pow(one_m, 1.0 / 1.33);
  sbl[b] = s_bl;
  ssl[b] = s_sl;
  __syncthreads();
  for (int s = 32; s > 0; s >>= 1) {
    if (b < s) {
      sterm[b] += sterm[b + s];
      sbl[b] += sbl[b + s];
      ssl[b] += ssl[b + s];
    }
    __syncthreads();
  }
  if (b == 0) {
    const double N = 64.0 * 512.0 * 512.0;
    out[0] = (float)(sterm[0] / 64.0 + 0.5 * (sbl[0] / N) + 0.3 * (ssl[0] / N));
  }
}

extern "C" void kernel_launch(void* const* d_in, const int* in_sizes, int n_in,
                              void* d_out, int out_size, void* d_ws,
                              size_t ws_size, hipStream_t stream) {
  const float* pred = (const float*)d_in[0];
  const float* target = (const float*)d_in[1];
  float* parts = (float*)d_ws;  // 64*16 tiles * 5 floats = 20,480 B
  dim3 grid(16, 64);
  polyp_main<<<grid, 256, 0, stream>>>(pred, target, parts);
  polyp_final<<<1, 64, 0, stream>>>(parts, (float*)d_out);
}